// LocalSpatialEncoding_68856915689534
// MI455X (gfx1250) — compile-verified
//
#include <hip/hip_runtime.h>

typedef float v2f __attribute__((ext_vector_type(2)));
typedef float v4f __attribute__((ext_vector_type(4)));
typedef float v8f __attribute__((ext_vector_type(8)));

namespace {
constexpr int BATCH = 4;
constexpr int NPTS  = 40960;
constexpr int KNN   = 16;
constexpr int DCH   = 16;
constexpr float EPS = 1e-5f;
constexpr int WAVES_PER_BLOCK = 8;   // 256 threads / wave32
constexpr int PPW = 8;               // points per wave (NPTS % PPW == 0)
}

// Single-instruction ReLU: avoids clang's canonicalize+max pair.
__device__ __forceinline__ float relu1(float x) {
    float r;
    asm("v_max_num_f32 %0, 0, %1" : "=v"(r) : "v"(x));
    return r;
}

// One wave32 handles PPW consecutive points (b, n0..n0+PPW-1).
// Per point: D(16x16) = A(16x12 fused conv*bn) x S(12x16 spatial), via three
// accumulating V_WMMA_F32_16X16X4_F32. Channels padded: c10 = 1.0 (bias), c11 = 0.
// D layout: vgpr v, lane l -> M = v + 8*(l>=16), N = l%16  ==> each D VGPR is one
// output channel row, 64B contiguous over k, matching out[b,oc,n,0..15].
__global__ __launch_bounds__(256) void lse_wmma_kernel(
    const float* __restrict__ coords,    // [B,N,3]
    const float* __restrict__ features,  // [B,D,N]
    const int*   __restrict__ idx,       // [B,N,K]
    const float* __restrict__ dist,      // [B,N,K]
    const float* __restrict__ conv_w,    // [D,10]
    const float* __restrict__ conv_b,    // [D]
    const float* __restrict__ bn_gamma,
    const float* __restrict__ bn_beta,
    const float* __restrict__ bn_mean,
    const float* __restrict__ bn_var,
    float* __restrict__ out)             // [B, 2D, N, K]
{
    const int lane = threadIdx.x & 31;
    const int k    = lane & 15;          // neighbor column
    const int half = lane >> 4;          // 0: lanes 0-15, 1: lanes 16-31

    int wave = blockIdx.x * WAVES_PER_BLOCK + (threadIdx.x >> 5);
    wave = __builtin_amdgcn_readfirstlane(wave);       // uniform -> SALU/s_load paths
    const int p0 = wave * PPW;
    const int b  = p0 / NPTS;                          // range never crosses batch
    const int n0 = p0 - b * NPTS;

    // ---- A matrix: BN folded into conv weights; built ONCE per wave ----
    const int o = k;                                   // A row = output channel
    const float inv = bn_gamma[o] * rsqrtf(bn_var[o] + EPS);
    const float bp  = (conv_b[o] - bn_mean[o]) * inv + bn_beta[o];
    const float* wrow = conv_w + o * 10;

    v2f A0, A1, A2;
    A0[0] = inv * wrow[0 + 2 * half];  A0[1] = inv * wrow[1 + 2 * half];   // c 0,1 / 2,3
    A1[0] = inv * wrow[4 + 2 * half];  A1[1] = inv * wrow[5 + 2 * half];   // c 4,5 / 6,7
    A2[0] = half ? bp   : inv * wrow[8];                                   // c 8 / 10(bias)
    A2[1] = half ? 0.0f : inv * wrow[9];                                   // c 9 / 11(pad)

    // ---- per-wave base pointers (per-point step folds into VMEM immediates) ----
    const size_t pb0 = (size_t)b * NPTS + n0;
    const size_t CH  = (size_t)NPTS * KNN;                       // out channel stride
    const float* cbp = coords + pb0 * 3;                         // uniform base coords
    const int*   ip  = idx    + pb0 * KNN + k;
    const float* dp  = dist   + pb0 * KNN + k;
    float* op_s = out + (((size_t)b * 2 * DCH + 8 * half) * NPTS + n0) * KNN + k;

    // feature-broadcast lane remap: 4 lanes per channel row -> b128 stores
    const int drow = lane >> 2;                                  // 0..7
    const int kq   = (lane & 3) * 4;                             // 0,4,8,12
    const float* fpb = features + ((size_t)b * DCH + drow) * NPTS + n0;
    float* op_f0 = out + (((size_t)b * 2 * DCH + DCH + drow) * NPTS + n0) * KNN + kq;
    float* op_f1 = op_f0 + 8 * CH;                               // rows d=8..15 (>8MB imm)

#pragma unroll 2
    for (int p = 0; p < PPW; ++p) {
        // wave-uniform base point (scalarized)
        const float bx = cbp[3 * p + 0], by = cbp[3 * p + 1], bz = cbp[3 * p + 2];

        const int j = ip[p * KNN];                               // neighbor index
        const float* cn = coords + ((size_t)b * NPTS + j) * 3;   // L2-resident gather
        const float nx = cn[0], ny = cn[1], nz = cn[2];
        const float rx = bx - nx, ry = by - ny, rz = bz - nz;
        const float dd = dp[p * KNN];

        v2f S0, S1, S2;
        S0[0] = half ? bz   : bx;   S0[1] = half ? nx   : by;    // rows c0,c1 / c2,c3
        S1[0] = half ? rx   : ny;   S1[1] = half ? ry   : nz;    // rows c4,c5 / c6,c7
        S2[0] = half ? 1.0f : rz;   S2[1] = half ? 0.0f : dd;    // rows c8,c9 / c10,c11

        v8f acc = {};
        acc = __builtin_amdgcn_wmma_f32_16x16x4_f32(false, A0, false, S0, (short)0, acc, false, false);
        acc = __builtin_amdgcn_wmma_f32_16x16x4_f32(false, A1, false, S1, (short)0, acc, false, false);
        acc = __builtin_amdgcn_wmma_f32_16x16x4_f32(false, A2, false, S2, (short)0, acc, false, false);

        // spatial half: single-instruction ReLU + NT store; each VGPR = one channel row
#pragma unroll
        for (int v = 0; v < 8; ++v) {
            __builtin_nontemporal_store(relu1(acc[v]),
                                        op_s + (size_t)v * CH + (size_t)p * KNN);
        }

        // feature half: broadcast rows, 2x b128 NT stores cover all 16 channels
        const float f0 = fpb[p];                                 // d = lane>>2
        const float f1 = fpb[(size_t)8 * NPTS + p];              // d = lane>>2 + 8
        v4f w0; w0[0] = f0; w0[1] = f0; w0[2] = f0; w0[3] = f0;
        v4f w1; w1[0] = f1; w1[1] = f1; w1[2] = f1; w1[3] = f1;
        __builtin_nontemporal_store(w0, (v4f*)(op_f0 + (size_t)p * KNN));
        __builtin_nontemporal_store(w1, (v4f*)(op_f1 + (size_t)p * KNN));
    }
}

extern "C" void kernel_launch(void* const* d_in, const int* in_sizes, int n_in,
                              void* d_out, int out_size, void* d_ws, size_t ws_size,
                              hipStream_t stream) {
    const float* coords   = (const float*)d_in[0];
    const float* features = (const float*)d_in[1];
    const int*   idx      = (const int*)d_in[2];
    const float* dist     = (const float*)d_in[3];
    const float* conv_w   = (const float*)d_in[4];
    const float* conv_b   = (const float*)d_in[5];
    const float* bn_gamma = (const float*)d_in[6];
    const float* bn_beta  = (const float*)d_in[7];
    const float* bn_mean  = (const float*)d_in[8];
    const float* bn_var   = (const float*)d_in[9];
    float* out = (float*)d_out;

    const int total_waves = (BATCH * NPTS) / PPW;          // 20480 waves
    dim3 grid(total_waves / WAVES_PER_BLOCK);              // 2560 blocks, exact
    dim3 block(256);
    hipLaunchKernelGGL(lse_wmma_kernel, grid, block, 0, stream,
                       coords, features, idx, dist, conv_w, conv_b,
                       bn_gamma, bn_beta, bn_mean, bn_var, out);
}